// NpairLoss_74775380623881
// MI455X (gfx1250) — compile-verified
//
#include <hip/hip_runtime.h>
#include <math.h>

typedef __attribute__((ext_vector_type(2))) float v2f;
typedef __attribute__((ext_vector_type(8))) float v8f;

// D = [a1;a2] (16x512)  @  [p1;p2]^T (512x16), fp32 WMMA 16x16x4, K split over 4 waves.
// logit1 = D[0:8,0:8], logit2 = D[8:16,8:16]. Scalar CE epilogue on lane 0.
__global__ __launch_bounds__(128)
void npair_loss_kernel(const float* __restrict__ emb,   // (32,512)
                       const float* __restrict__ n2,    // (32,2)
                       const float* __restrict__ n4,    // (32,8)
                       const int*   __restrict__ tgt,   // (32,)
                       float* __restrict__ out)         // scalar
{
    __shared__ float parts[4][16][16];   // per-wave partial Gram tiles
    __shared__ float sqp[128][2];        // per-thread partial sums of squares

    const int tid  = threadIdx.x;
    const int wave = tid >> 5;
    const int lane = tid & 31;
    const int half = lane >> 4;          // K-half selector within fragment
    const int mn   = lane & 15;          // M (for A) / N (for B)

    // Row mapping: A rows 0-7 -> emb[4j] (i11), rows 8-15 -> emb[4j+2] (i21)
    //              B rows 0-7 -> emb[4j+1] (i12), rows 8-15 -> emb[4j+3] (i22)
    const int arow = (mn < 8) ? (4 * mn)     : (4 * (mn - 8) + 2);
    const int brow = (mn < 8) ? (4 * mn + 1) : (4 * (mn - 8) + 3);
    const float* aBase = emb + arow * 512 + 2 * half;  // lanes 0-15: K={0,1}; 16-31: K={2,3}
    const float* bBase = emb + brow * 512 + 2 * half;

    v8f c = {};
    const int k0 = wave * 128;           // each wave owns a K-quarter
    #pragma unroll
    for (int k = 0; k < 128; k += 4) {
        v2f a = *(const v2f*)(aBase + k0 + k);
        v2f b = *(const v2f*)(bBase + k0 + k);
        // (neg_a, A, neg_b, B, c_mod, C, reuse_a, reuse_b)
        c = __builtin_amdgcn_wmma_f32_16x16x4_f32(false, a, false, b,
                                                  (short)0, c, false, false);
    }
    // C/D layout: VGPR r -> M = r + 8*half, N = mn
    #pragma unroll
    for (int r = 0; r < 8; ++r)
        parts[wave][r + half * 8][mn] = c[r];

    // ---- L2 term partials: rows with r%4 in {0,1} -> set1, {2,3} -> set2 ----
    float s1 = 0.f, s2 = 0.f;
    for (int idx = tid; idx < 32 * 512; idx += 128) {
        float v = emb[idx];
        if (((idx >> 9) & 3) < 2) s1 += v * v; else s2 += v * v;
    }
    sqp[tid][0] = s1;
    sqp[tid][1] = s2;

    __syncthreads();
    if (tid != 0) return;

    // ---- deterministic reductions ----
    float D[16][16];
    for (int m = 0; m < 16; ++m)
        for (int n = 0; n < 16; ++n)
            D[m][n] = parts[0][m][n] + parts[1][m][n] + parts[2][m][n] + parts[3][m][n];
    float sq1 = 0.f, sq2 = 0.f;
    for (int i = 0; i < 128; ++i) { sq1 += sqp[i][0]; sq2 += sqp[i][1]; }

    // ---- scalar epilogue (exact reference math) ----
    int  t8[8]; bool pol[8]; int pid[8];
    for (int j = 0; j < 8; ++j) {
        t8[j]  = tgt[4 * j];
        pol[j] = (t8[j] < 4);
        pid[j] = pol[j] ? 1 : 0;
    }

    float lg1[8][8], lg2[8][8];
    for (int r = 0; r < 8; ++r) {
        for (int q = 0; q < 8; ++q) {
            float w1 = (pol[r] == pol[q]) ? 1.f
                     : expf(n2[(4 * r) * 2 + pid[q]] + n2[(4 * q + 1) * 2 + pid[r]]);
            float e1 = (r == q) ? 1.f
                     : expf(n4[(4 * r) * 8 + t8[q]] + n4[(4 * q + 1) * 8 + t8[r]]);
            lg1[r][q] = D[r][q] * w1 * e1;

            float w2 = (pol[r] == pol[q]) ? 1.f
                     : expf(n2[(4 * r + 2) * 2 + pid[q]] + n2[(4 * q + 3) * 2 + pid[r]]);
            float e2 = (r == q) ? 1.f
                     : expf(n4[(4 * r + 2) * 8 + t8[q]] + n4[(4 * q + 3) * 8 + t8[r]]);
            lg2[r][q] = D[8 + r][8 + q] * w2 * e2;
        }
    }

    int label[8];
    for (int j = 0; j < 8; ++j) label[t8[j]] = j;   // inverse permutation
    int idx1[4], idx2[4];
    for (int j = 0; j < 4; ++j) { idx1[j] = label[j]; idx2[j] = label[4 + j]; }

    float total_ce = 0.f;
    float (*mats[2])[8] = { lg1, lg2 };
    for (int m = 0; m < 2; ++m) {
        float (*lg)[8] = mats[m];

        // CE over the 8x4 "lossp" rows, one-hot at col r%4
        float ce3 = 0.f;
        for (int r = 0; r < 8; ++r) {
            const int* I = (r < 4) ? idx1 : idx2;
            int rr = I[r & 3];
            float row[4];
            for (int q = 0; q < 4; ++q) row[q] = lg[rr][I[q]];
            float mx = row[0];
            for (int q = 1; q < 4; ++q) mx = fmaxf(mx, row[q]);
            float s = 0.f;
            for (int q = 0; q < 4; ++q) s += expf(row[q] - mx);
            ce3 -= row[r & 3] - mx - logf(s);
        }
        ce3 /= 8.f;

        // CE over the (c0,c1) 2-logit rows, one-hot at col 0
        float ce2 = 0.f;
        for (int r = 0; r < 8; ++r) {
            float sA = 0.f, sB = 0.f;
            for (int q = 0; q < 4; ++q) { sA += lg[r][idx1[q]]; sB += lg[r][idx2[q]]; }
            float dd = lg[r][r];
            float c0 = pol[r] ? (sA - dd) / 3.0f : (sB - dd) / 3.0f;
            float c1 = pol[r] ? sB / 4.0f        : sA / 4.0f;
            float mx = fmaxf(c0, c1);
            float s  = expf(c0 - mx) + expf(c1 - mx);
            ce2 -= c0 - mx - logf(s);
        }
        ce2 /= 8.f;

        total_ce += 0.5f * ce2 + 0.5f * ce3;
    }

    const float L2_REG = 0.02f;
    float l2_1  = sq1 / 8.0f;   // sum a1^2 + sum p1^2, both over /bs=8
    float l2_2r = sq2 / 8.0f;
    out[0] = (total_ce + L2_REG * l2_1 * 0.25f + L2_REG * l2_2r * 0.25f) * 0.5f;
}

extern "C" void kernel_launch(void* const* d_in, const int* in_sizes, int n_in,
                              void* d_out, int out_size, void* d_ws, size_t ws_size,
                              hipStream_t stream) {
    const float* emb = (const float*)d_in[0];
    const float* n2  = (const float*)d_in[1];
    const float* n4  = (const float*)d_in[2];
    const int*   tgt = (const int*)d_in[3];
    float* out = (float*)d_out;
    hipLaunchKernelGGL(npair_loss_kernel, dim3(1), dim3(128), 0, stream,
                       emb, n2, n4, tgt, out);
}